// MultiLanguageCodeEmbedding_75076028334454
// MI455X (gfx1250) — compile-verified
//
#include <hip/hip_runtime.h>
#include <hip/hip_bf16.h>

typedef __attribute__((ext_vector_type(2))) float v2f;
typedef __attribute__((ext_vector_type(8))) float v8f;

#define Hc 512
#define Rc 64
#define SV 25000
#define Sc 2048
#define Bc 32

__device__ __forceinline__ v8f wmma4(v2f a, v2f b, v8f c) {
    // D = A(16x4,f32) * B(4x16,f32) + C(16x16,f32)
    return __builtin_amdgcn_wmma_f32_16x16x4_f32(
        /*neg_a=*/false, a, /*neg_b=*/false, b,
        /*c_mod=*/(short)0, c, /*reuse_a=*/false, /*reuse_b=*/false);
}

__device__ __forceinline__ int imax(int a, int b) { return a > b ? a : b; }

// ---------------------------------------------------------------------------
// Kernel 1: per-row clamped-scan (nesting) + prefix-sum mod 8 (segments).
// One wave (32 lanes) per batch row; chunk = 64 tokens/lane; shuffle scan of
// the composed function f(c) = max(m, c + s), which is associative.
// ---------------------------------------------------------------------------
__global__ __launch_bounds__(32) void syntax_scan(const int* __restrict__ tok,
                                                  int* __restrict__ nest,
                                                  int* __restrict__ seg) {
    const int b = blockIdx.x;
    const int lane = threadIdx.x;
    const int CH = Sc / 32;  // 64
    const int base = b * Sc + lane * CH;

    int m = -1000000000, sdelta = 0, gsum = 0;
    for (int i = 0; i < CH; ++i) {
        int t = tok[base + i];
        int d = ((t == 40) | (t == 123) | (t == 91)) -
                ((t == 41) | (t == 125) | (t == 93));
        m = imax(0, m + d);
        sdelta += d;
        gsum += (t > 39990);
    }
    // inclusive scan across lanes: combine(left,right): m=max(mr, ml+sr), s=sl+sr
    int Mi = m, Si = sdelta, Gi = gsum;
    for (int o = 1; o < 32; o <<= 1) {
        int Mo = __shfl_up(Mi, o, 32);
        int So = __shfl_up(Si, o, 32);
        int Go = __shfl_up(Gi, o, 32);
        if (lane >= o) { Mi = imax(Mi, Mo + Si); Si = So + Si; Gi = Go + Gi; }
    }
    int Mex = __shfl_up(Mi, 1, 32);
    int Sex = __shfl_up(Si, 1, 32);
    int Gex = __shfl_up(Gi, 1, 32);
    int carry = 0, gcarry = 0;
    if (lane > 0) { carry = imax(Mex, Sex); gcarry = Gex; }

    int c = carry, g = gcarry;
    for (int i = 0; i < CH; ++i) {
        int t = tok[base + i];
        int d = ((t == 40) | (t == 123) | (t == 91)) -
                ((t == 41) | (t == 125) | (t == 93));
        c = imax(0, c + d);
        nest[base + i] = c < 15 ? c : 15;
        g += (t > 39990);
        seg[base + i] = g & 7;
    }
}

// ---------------------------------------------------------------------------
// Kernel 2: NN[16][512] = nesting_table @ W1^T ; SS[8][512] = segment_table @ W2^T
// proj_w is (H, 3H) row-major; W1 cols 512..1023, W2 cols 1024..1535.
// ---------------------------------------------------------------------------
__global__ __launch_bounds__(256) void small_proj(const float* __restrict__ ntab,
                                                  const float* __restrict__ stab,
                                                  const float* __restrict__ pw,
                                                  float* __restrict__ NNo,
                                                  float* __restrict__ SSo) {
    const int r = blockIdx.x;  // 0..23
    const float* src;
    float* dst;
    int off;
    if (r < 16) { src = ntab + r * Hc; dst = NNo + r * Hc; off = 512; }
    else        { src = stab + (r - 16) * Hc; dst = SSo + (r - 16) * Hc; off = 1024; }
    for (int h = threadIdx.x; h < Hc; h += 256) {
        float acc = 0.f;
        const float* wr = pw + (size_t)h * 1536 + off;
        for (int k = 0; k < Hc; ++k) acc = fmaf(src[k], wr[k], acc);
        dst[h] = acc;
    }
}

// ---------------------------------------------------------------------------
// Kernel 3: PP[2048][512] = pos_table @ W0^T via f32 WMMA. One 16x16 tile/wave.
// ---------------------------------------------------------------------------
__global__ __launch_bounds__(128) void pos_proj(const float* __restrict__ pos,
                                                const float* __restrict__ pw,
                                                float* __restrict__ PPo) {
    const int lane = threadIdx.x & 31;
    const int tile = blockIdx.x * 4 + (threadIdx.x >> 5);  // 4096 tiles
    const int s0 = (tile >> 5) * 16;   // 128 s-tiles
    const int h0 = (tile & 31) * 16;   // 32 h-tiles
    const int mA = lane & 15;
    const int kb = (lane >> 4) << 1;
    const int n = lane & 15;

    const float* ap = pos + (size_t)(s0 + mA) * Hc + kb;
    const float* bp = pw + (size_t)(h0 + n) * 1536 + kb;  // W0 cols 0..511
    v8f acc = {};
    for (int k0 = 0; k0 < Hc; k0 += 4) {
        v2f a = *(const v2f*)(ap + k0);
        v2f bb = *(const v2f*)(bp + k0);
        acc = wmma4(a, bb, acc);
    }
    const int rbase = (lane >> 4) << 3;
#pragma unroll
    for (int i = 0; i < 8; ++i)
        PPo[(size_t)(s0 + rbase + i) * Hc + h0 + n] = acc[i];
}

// ---------------------------------------------------------------------------
// Kernel 4: main fused embedding. One wave per 16-token tile.
//   GEMM1: low(16x64) = shared_emb(16x512) @ dw[lang]^T  (WMMA f32, K=512)
//   stage low -> LDS (stride 66 floats to avoid bank conflicts)
//   GEMM2: lora(16x512) = low @ uw[lang]^T (K=64), fused epilogue + store
// ---------------------------------------------------------------------------
__global__ __launch_bounds__(256) void emb_kernel(
    const int* __restrict__ tok, const int* __restrict__ langids,
    const int* __restrict__ ttypes, const float* __restrict__ shared_table,
    const float* __restrict__ adown, const float* __restrict__ aup,
    const float* __restrict__ lang_tables, const float* __restrict__ ltt,
    const float* __restrict__ ttt, const float* __restrict__ PP,
    const float* __restrict__ NN, const float* __restrict__ SS,
    const int* __restrict__ nest, const int* __restrict__ seg,
    float* __restrict__ out) {
    __shared__ float lowbuf[8][16 * 66];

    const int wave = threadIdx.x >> 5;
    const int lane = threadIdx.x & 31;
    const int tile = blockIdx.x * 8 + wave;   // 4096 tiles total
    const int b = tile >> 7;                  // 128 tiles per batch row
    const int s0 = (tile & 127) << 4;

    const int lang = langids[b];
    const float* dw = adown + (size_t)lang * (Rc * Hc);        // (64, 512)
    const float* uw = aup + (size_t)lang * (Hc * Rc);          // (512, 64)
    const float* lt = lang_tables + (size_t)lang * (2500 * Hc);

    const int mA = lane & 15;
    const int kb = (lane >> 4) << 1;
    const int n = lane & 15;
    const int rbase = (lane >> 4) << 3;

    // A-row pointer for GEMM1 (this lane's token row, clamped shared id)
    {
        // nothing
    }
    int tA = tok[b * Sc + s0 + mA];
    const float* arow = shared_table + (size_t)(tA < SV ? tA : 0) * Hc + kb;
    const float* dwn = dw + (size_t)n * Hc + kb;  // B row n base (k contiguous)

    v8f a0 = {}, a1 = {}, a2 = {}, a3 = {};
    for (int k0 = 0; k0 < Hc; k0 += 4) {
        v2f a = *(const v2f*)(arow + k0);
        v2f b0 = *(const v2f*)(dwn + k0);
        v2f b1 = *(const v2f*)(dwn + 16 * Hc + k0);
        v2f b2 = *(const v2f*)(dwn + 32 * Hc + k0);
        v2f b3 = *(const v2f*)(dwn + 48 * Hc + k0);
        a0 = wmma4(a, b0, a0);
        a1 = wmma4(a, b1, a1);
        a2 = wmma4(a, b2, a2);
        a3 = wmma4(a, b3, a3);
    }
    float* lw = &lowbuf[wave][0];
#pragma unroll
    for (int i = 0; i < 8; ++i) {
        int row = rbase + i;
        lw[row * 66 + 0 + n] = a0[i];
        lw[row * 66 + 16 + n] = a1[i];
        lw[row * 66 + 32 + n] = a2[i];
        lw[row * 66 + 48 + n] = a3[i];
    }
    __syncthreads();  // order LDS stores vs cross-lane reads (uniform control flow)

    // Per-lane epilogue metadata for rows rbase+0..7
    int sidE[8], lidE[8], ttE[8], nsE[8], sgE[8];
#pragma unroll
    for (int i = 0; i < 8; ++i) {
        int row = rbase + i;
        int t = tok[b * Sc + s0 + row];
        sidE[i] = (t < SV) ? t : 0;
        lidE[i] = (t < SV) ? 0 : (t - SV);
        ttE[i] = ttypes[b * Sc + s0 + row];
        nsE[i] = nest[b * Sc + s0 + row];
        sgE[i] = seg[b * Sc + s0 + row];
    }

    const float sqrtH = 22.62741699796952f;  // sqrt(512)
    const float* alw = lw + mA * 66 + kb;
    for (int hr = 0; hr < 32; ++hr) {
        const int h0 = hr * 16;
        const float* bp = uw + (size_t)(h0 + n) * Rc + kb;
        v8f acc = {};
#pragma unroll
        for (int k0 = 0; k0 < Rc; k0 += 4) {
            v2f a = *(const v2f*)(alw + k0);
            v2f bb = *(const v2f*)(bp + k0);
            acc = wmma4(a, bb, acc);
        }
        const int h = h0 + n;
        const float ltv = ltt[lang * Hc + h];
#pragma unroll
        for (int i = 0; i < 8; ++i) {
            int s = s0 + rbase + i;
            float v = acc[i] * (1.0f / 64.0f)
                    + shared_table[(size_t)sidE[i] * Hc + h]
                    + lt[(size_t)lidE[i] * Hc + h]
                    + ltv
                    + ttt[(size_t)ttE[i] * Hc + h]
                    + PP[(size_t)s * Hc + h]
                    + NN[(size_t)nsE[i] * Hc + h]
                    + SS[(size_t)sgE[i] * Hc + h];
            out[((size_t)b * Sc + s) * Hc + h] = v * sqrtH;
        }
    }
}

extern "C" void kernel_launch(void* const* d_in, const int* in_sizes, int n_in,
                              void* d_out, int out_size, void* d_ws, size_t ws_size,
                              hipStream_t stream) {
    (void)in_sizes; (void)n_in; (void)out_size; (void)ws_size;
    const int* tok          = (const int*)d_in[0];
    const int* langids      = (const int*)d_in[1];
    const int* ttypes       = (const int*)d_in[2];
    const float* shared_tab = (const float*)d_in[3];
    const float* adown      = (const float*)d_in[4];
    const float* aup        = (const float*)d_in[5];
    const float* lang_tabs  = (const float*)d_in[6];
    const float* ltt        = (const float*)d_in[7];
    const float* ttt        = (const float*)d_in[8];
    const float* pos_tab    = (const float*)d_in[9];
    const float* ntab       = (const float*)d_in[10];
    const float* stab       = (const float*)d_in[11];
    const float* pw         = (const float*)d_in[12];
    float* out = (float*)d_out;

    // Workspace layout
    float* PPo = (float*)d_ws;                 // 2048*512
    float* NNo = PPo + (size_t)Sc * Hc;        // 16*512
    float* SSo = NNo + 16 * Hc;                // 8*512
    int* nest  = (int*)(SSo + 8 * Hc);         // B*S
    int* seg   = nest + (size_t)Bc * Sc;       // B*S

    syntax_scan<<<Bc, 32, 0, stream>>>(tok, nest, seg);
    small_proj<<<24, 256, 0, stream>>>(ntab, stab, pw, NNo, SSo);
    pos_proj<<<1024, 128, 0, stream>>>(pos_tab, pw, PPo);
    emb_kernel<<<512, 256, 0, stream>>>(tok, langids, ttypes, shared_tab, adown,
                                        aup, lang_tabs, ltt, ttt, PPo, NNo, SSo,
                                        nest, seg, out);
}